// ContrastiveLoss_V4_80410377716009
// MI455X (gfx1250) — compile-verified
//
#include <hip/hip_runtime.h>

// ---------------------------------------------------------------------------
// Contrastive loss, MI455X (gfx1250, wave32).
// Roofline: gram = e@e^T is 34 GFLOP (V_WMMA_F32_16X16X4_F32, fp32-exact),
// but the per-pair epilogue is the issue bottleneck (268M pairs, VALU-bound).
// The 4MB embedding matrix is L2-resident; HBM traffic ~ nil.
// This round: the negative-comparison COUNT is removed from the hot loop —
// it equals N^2 - sum_c(count_c^2), computed once by a tiny label-histogram
// kernel. Saves cndmask+add per pair (~20% of the remaining epilogue VALU).
// Retained: raw v_sqrt_f32, packed pk_fma epilogue, async double-buffered LDS
// panels (global_load_async_to_lds_b128 / s_wait_asynccnt).
// ---------------------------------------------------------------------------

typedef __attribute__((ext_vector_type(2))) float v2f;
typedef __attribute__((ext_vector_type(8))) float v8f;
typedef __attribute__((address_space(3))) float lds_f;

#define N_TOT   16384
#define D_DIM   64
#define C_CLS   128                 // label cardinality (labels in [0, C_CLS))
#define MARGIN_ 1.0f
#define EPS_NORM_ 1e-6f
#define EPS_PD_   1e-6f

#define ROWS_PER_BLOCK 128          // 8 waves * 16 rows
#define COLS_PER_CHUNK 128          // LDS panel width
#define COL_SLICES     8            // gridDim.y
#define PANEL_STRIDE   68           // 16B-aligned rows, bank-conflict-free b64 reads

static_assert(N_TOT % ROWS_PER_BLOCK == 0, "");
static_assert((N_TOT / COL_SLICES) % COLS_PER_CHUNK == 0, "");

// Raw-hardware sqrt: argument always finite, positive, well-scaled here.
__device__ __forceinline__ float fast_sqrtf(float x) {
#if __has_builtin(__builtin_amdgcn_sqrtf)
  return __builtin_amdgcn_sqrtf(x);
#else
  return sqrtf(x);
#endif
}

__device__ __forceinline__ float waveReduceSumF(float v) {
#pragma unroll
  for (int m = 16; m >= 1; m >>= 1) v += __shfl_xor(v, m, 32);
  return v;
}

// ---- gfx1250 async global->LDS copy (ASYNCcnt-tracked) --------------------
__device__ __forceinline__ unsigned lds_byte_off(const float* p) {
  // generic -> AS(3) addrspacecast; AS(3) pointer value == LDS byte offset
  return (unsigned)(unsigned long long)(lds_f*)(void*)const_cast<float*>(p);
}
__device__ __forceinline__ void async_copy_b128(unsigned lds_dst, const float4* src) {
  asm volatile("global_load_async_to_lds_b128 %0, %1, off"
               :: "v"(lds_dst), "v"(src) : "memory");
}
__device__ __forceinline__ void wait_async0() {
  asm volatile("s_wait_asynccnt 0x0" ::: "memory");
}

// --------------------------- init / finalize -------------------------------
__global__ void cl_init_kernel(double* acc) {
  acc[0] = 0.0;   // pos_loss
  acc[1] = 0.0;   // neg_loss
}

__global__ void cl_fin_kernel(const double* acc, const unsigned int* cnt, float* out) {
  double total = acc[0] + acc[1];
  double ncomp = (double)N_TOT + (double)(*cnt);
  out[0] = (float)(total / ncomp);
}

// ------------- label histogram -> exact negative-comparison count ----------
// sum(neg_mask) = N^2 - sum_c count_c^2  (diagonal excluded automatically:
// i==j always has equal labels). Single block, O(N), negligible cost.
__global__ void __launch_bounds__(256)
cl_hist_kernel(const int* __restrict__ lab, unsigned int* __restrict__ cnt) {
  __shared__ unsigned int bins[C_CLS];
  const int tid = threadIdx.x;
  for (int c = tid; c < C_CLS; c += 256) bins[c] = 0u;
  __syncthreads();
  for (int i = tid; i < N_TOT; i += 256) atomicAdd(&bins[lab[i]], 1u);
  __syncthreads();
  if (tid == 0) {
    unsigned int same = 0u;
    for (int c = 0; c < C_CLS; ++c) same += bins[c] * bins[c];
    *cnt = (unsigned int)((unsigned long long)N_TOT * N_TOT - same);
  }
}

// --------------------------- row normalization -----------------------------
__global__ void __launch_bounds__(256)
cl_norm_kernel(const float* __restrict__ emb, float* __restrict__ e,
               float* __restrict__ sq, float* __restrict__ s) {
  const int wave = threadIdx.x >> 5;
  const int lane = threadIdx.x & 31;
  const int row  = blockIdx.x * 8 + wave;
  const float2 v = *(const float2*)(emb + row * D_DIM + lane * 2);
  float tss = waveReduceSumF(v.x * v.x + v.y * v.y);
  float tsm = waveReduceSumF(v.x + v.y);
  float nrm = fast_sqrtf(tss);
  float inv = 1.0f / fmaxf(nrm, EPS_NORM_);
  float2 o; o.x = v.x * inv; o.y = v.y * inv;
  *(float2*)(e + row * D_DIM + lane * 2) = o;
  if (lane == 0) {
    sq[row] = tss * inv * inv;
    s[row]  = tsm * inv;
  }
}

// --------------------------- main pairwise kernel --------------------------
__global__ void __launch_bounds__(256)
cl_tile_kernel(const float* __restrict__ e, const float* __restrict__ sq,
               const float* __restrict__ s, const int* __restrict__ lab,
               double* __restrict__ acc_neg) {
  __shared__ float panel[2][COLS_PER_CHUNK * PANEL_STRIDE];   // 68 KB

  const int tid  = threadIdx.x;
  const int wave = tid >> 5;
  const int lane = tid & 31;
  const int m16  = lane & 15;
  const int hi   = lane >> 4;        // 0: lanes 0-15, 1: lanes 16-31
  const int ksel = hi << 1;          // K offset 0 or 2 within each K=4 step

  const int rowWave = blockIdx.x * ROWS_PER_BLOCK + wave * 16;
  const int colSliceBase = blockIdx.y * (N_TOT / COL_SLICES);

  // ---- A fragments: wave's 16 rows, all 16 K-steps, kept in registers ----
  // ISA 32-bit A 16x4 layout: lanes 0-15 hold K{0,1}; lanes 16-31 hold K{2,3}.
  v2f afrag[16];
  const float* arow = e + (size_t)(rowWave + m16) * D_DIM;
#pragma unroll
  for (int k = 0; k < 16; ++k) {
    float2 t = *(const float2*)(arow + 4 * k + ksel);
    afrag[k] = (v2f){t.x, t.y};
  }

  // ---- row-side hoisted epilogue bases per accumulator register ----
  // C/D layout: acc VGPR v holds M = v (lanes 0-15) / M = v+8 (lanes 16-31).
  const float Deps2 = (float)D_DIM * EPS_PD_ * EPS_PD_;
  float basem[8];
  int   labm[8];
  const int mofs = hi * 8;
#pragma unroll
  for (int v = 0; v < 8; ++v) {
    const int m = rowWave + mofs + v;
    basem[v] = fmaf(2.0f * EPS_PD_, s[m], sq[m]) + Deps2;  // sqm + 2eps*sm + D*eps^2
    labm[v]  = lab[m];
  }

  float negAcc = 0.0f;

  const int nChunks = (N_TOT / COL_SLICES) / COLS_PER_CHUNK;

  // ---- async panel issue (each thread: 1 column half, 8x b128) ----
  const int pc    = tid >> 1;
  const int phalf = (tid & 1) * 32;
  const unsigned pdst0 = lds_byte_off(&panel[0][pc * PANEL_STRIDE + phalf]);
  const unsigned pdst1 = lds_byte_off(&panel[1][pc * PANEL_STRIDE + phalf]);

  // prologue: issue chunk 0 into buffer 0
  {
    const float4* src = (const float4*)(e + (size_t)(colSliceBase + pc) * D_DIM + phalf);
#pragma unroll
    for (int j = 0; j < 8; ++j) async_copy_b128(pdst0 + j * 16, src + j);
  }

  for (int chunk = 0; chunk < nChunks; ++chunk) {
    const int colBase = colSliceBase + chunk * COLS_PER_CHUNK;
    const int buf = chunk & 1;

    wait_async0();      // my wave's copies for this chunk are in LDS
    __syncthreads();    // everyone's copies are in LDS

    // overlap: issue next chunk's copies into the other buffer
    if (chunk + 1 < nChunks) {
      const int nb = colBase + COLS_PER_CHUNK;
      const float4* src = (const float4*)(e + (size_t)(nb + pc) * D_DIM + phalf);
      const unsigned dst = (buf ? pdst0 : pdst1);
#pragma unroll
      for (int j = 0; j < 8; ++j) async_copy_b128(dst + j * 16, src + j);
    }

#pragma unroll
    for (int tc = 0; tc < COLS_PER_CHUNK / 16; ++tc) {
      const int nloc = tc * 16 + m16;
      const float* bcol = &panel[buf][nloc * PANEL_STRIDE];

      v8f acc = {0.f, 0.f, 0.f, 0.f, 0.f, 0.f, 0.f, 0.f};
#pragma unroll
      for (int k = 0; k < 16; ++k) {
        float2 t = *(const float2*)(bcol + 4 * k + ksel);
        v2f b = (v2f){t.x, t.y};
        acc = __builtin_amdgcn_wmma_f32_16x16x4_f32(
            /*neg_a=*/false, afrag[k], /*neg_b=*/false, b,
            /*c_mod=*/(short)0, acc, /*reuse_a=*/false, /*reuse_b=*/false);
      }

      // column-side meta straight from L2-resident globals
      const int ncol = colBase + nloc;
      const float sqn = sq[ncol];
      const float sn  = s[ncol];
      const int   ln  = lab[ncol];
      const float basen = fmaf(-2.0f * EPS_PD_, sn, sqn);  // sqn - 2eps*sn
#pragma unroll
      for (int v = 0; v < 8; ++v) {
        float d2 = fmaf(-2.0f, acc[v], basem[v] + basen);
        float dist = fast_sqrtf(fmaxf(d2, 1e-12f));
        if (labm[v] != ln) {
          float h = fmaxf(MARGIN_ - dist, 0.0f);
          negAcc += h * h;
        }
      }
    }
  }

  negAcc = waveReduceSumF(negAcc);
  if (lane == 0) atomicAdd(acc_neg, (double)negAcc);
}

// --------------------------- positive-pair term ----------------------------
__global__ void __launch_bounds__(256)
cl_pos_kernel(const float* __restrict__ e, const float* __restrict__ sq,
              const float* __restrict__ s, const int* __restrict__ pos_idx,
              double* __restrict__ acc_pos) {
  const int i = blockIdx.x * blockDim.x + threadIdx.x;
  const int p = pos_idx[i];
  const float4* a = (const float4*)(e + (size_t)i * D_DIM);
  const float4* b = (const float4*)(e + (size_t)p * D_DIM);
  float dot = 0.0f;
#pragma unroll
  for (int k = 0; k < D_DIM / 4; ++k) {
    float4 x = a[k], y = b[k];
    dot = fmaf(x.x, y.x, dot);
    dot = fmaf(x.y, y.y, dot);
    dot = fmaf(x.z, y.z, dot);
    dot = fmaf(x.w, y.w, dot);
  }
  float d2 = sq[i] + sq[p] - 2.0f * dot
           + 2.0f * EPS_PD_ * (s[i] - s[p])
           + (float)D_DIM * EPS_PD_ * EPS_PD_;
  d2 = fmaxf(d2, 0.0f);
  float t = waveReduceSumF(d2);
  if ((threadIdx.x & 31) == 0) atomicAdd(acc_pos, (double)t);
}

// --------------------------- launch ----------------------------------------
extern "C" void kernel_launch(void* const* d_in, const int* in_sizes, int n_in,
                              void* d_out, int out_size, void* d_ws, size_t ws_size,
                              hipStream_t stream) {
  const float* emb     = (const float*)d_in[0];
  const int*   labels  = (const int*)d_in[1];
  const int*   pos_idx = (const int*)d_in[2];
  float*       out     = (float*)d_out;

  // workspace layout
  float* e  = (float*)d_ws;                 // N*D normalized embeddings
  float* sq = e + (size_t)N_TOT * D_DIM;    // per-row sum(e*e)
  float* s  = sq + N_TOT;                   // per-row sum(e)
  double* acc = (double*)(s + N_TOT);       // [0]=pos_loss, [1]=neg_loss (8B aligned)
  unsigned int* cnt = (unsigned int*)(acc + 2);

  cl_init_kernel<<<1, 1, 0, stream>>>(acc);
  cl_hist_kernel<<<1, 256, 0, stream>>>(labels, cnt);
  cl_norm_kernel<<<N_TOT / 8, 256, 0, stream>>>(emb, e, sq, s);
  {
    dim3 grid(N_TOT / ROWS_PER_BLOCK, COL_SLICES);
    cl_tile_kernel<<<grid, 256, 0, stream>>>(e, sq, s, labels, acc + 1);
  }
  cl_pos_kernel<<<N_TOT / 256, 256, 0, stream>>>(e, sq, s, pos_idx, acc + 0);
  cl_fin_kernel<<<1, 1, 0, stream>>>(acc, cnt, out);
}